// EnhancedTripletLoss_2293512536093
// MI455X (gfx1250) — compile-verified
//
#include <hip/hip_runtime.h>
#include <hip/hip_bf16.h>
#include <math.h>

// ---------------------------------------------------------------------------
// EnhancedTripletLoss for MI455X (gfx1250, wave32)
//   inputs : embeddings fp32 [8192, 256], labels int32 [8192]
//   output : scalar fp32 loss
// Fused WMMA Gram-tile mining; A tile resident in LDS, B chunks double-buffered
// and staged via GLOBAL_LOAD_ASYNC_TO_LDS_B128 (ASYNCcnt-tracked DMA).
// ---------------------------------------------------------------------------

typedef float v2f __attribute__((ext_vector_type(2)));
typedef float v8f __attribute__((ext_vector_type(8)));

#define DEMB   256   // embedding dim
#define TILE   128   // row/col tile
#define KB     128   // B K-chunk staged per pipeline stage
#define MARGIN 0.3f
#define PD_EPS 1e-6f

#define LDS_BYTES ((TILE * DEMB + 2 * TILE * KB) * 4)   // 128KB A + 2x64KB B = 256KB

// XOR swizzles: keep k,k+1 adjacent (b64 frag loads), spread banks across rows.
__device__ __forceinline__ int sw256(int row, int k) { return row * 256 + (k ^ ((row & 15) << 2)); }
__device__ __forceinline__ int sw128(int row, int k) { return row * 128 + (k ^ ((row & 15) << 2)); }

// Async DMA: 16B global -> LDS, per-lane LDS dest (swizzle-compatible), ASYNCcnt.
__device__ __forceinline__ void async_b128(unsigned lds_byte, const void* gaddr) {
    asm volatile("global_load_async_to_lds_b128 %0, %1, off"
                 :: "v"(lds_byte), "v"(gaddr) : "memory");
}
__device__ __forceinline__ void wait_async_le16() {
    asm volatile("s_wait_asynccnt 0x10" ::: "memory");
}
__device__ __forceinline__ void wait_async_le0() {
    asm volatile("s_wait_asynccnt 0x0" ::: "memory");
}

// ---------------------------------------------------------------------------
// Kernel 0: per-row squared norms; also zero the loss accumulators.
// ---------------------------------------------------------------------------
__global__ __launch_bounds__(256) void sqnorm_kernel(const float* __restrict__ X,
                                                     float* __restrict__ sq,
                                                     float* __restrict__ sumAcc,
                                                     unsigned* __restrict__ cntAcc) {
    const int a    = blockIdx.x * 8 + (threadIdx.x >> 5);
    const int lane = threadIdx.x & 31;
    const float* xa = X + (size_t)a * DEMB;
    float s = 0.f;
#pragma unroll
    for (int d0 = 0; d0 < DEMB / 32; ++d0) {
        float v = xa[lane + d0 * 32];
        s += v * v;
    }
#pragma unroll
    for (int off = 16; off >= 1; off >>= 1) s += __shfl_xor(s, off, 32);
    if (lane == 0) sq[a] = s;
    if (blockIdx.x == 0 && threadIdx.x == 0) { *sumAcc = 0.f; *cntAcc = 0u; }
}

// ---------------------------------------------------------------------------
// Kernel 1: fused Gram-tile (WMMA f32 16x16x4) + hard mining.
// One block = 128 anchor rows; 8 waves, each wave = 16 rows x 128 cols.
// ---------------------------------------------------------------------------
__global__ __launch_bounds__(256) void mine_kernel(const float* __restrict__ X,
                                                   const int* __restrict__ labels,
                                                   const float* __restrict__ sq,
                                                   int* __restrict__ posIdx,
                                                   int* __restrict__ negIdx,
                                                   int B) {
    extern __shared__ float smem[];
    float* As = smem;                         // 128 x 256, swizzled (resident)
    float* Bs = smem + TILE * DEMB;           // 2 x (128 x 128), swizzled

    const int tid     = threadIdx.x;
    const int wave    = tid >> 5;
    const int lane    = tid & 31;
    const int hf      = lane >> 4;            // half-wave select (M +8, K +2)
    const int l15     = lane & 15;
    const int rowBase = blockIdx.x * TILE;
    const int rw      = rowBase + wave * 16;
    const int NCHUNK  = 2 * (B / TILE);       // 128 B-chunks

    // --- issue A tile: 128x256 f32 = 8192 b128 slots, 32 per thread ---------
#pragma unroll
    for (int s = 0; s < 32; ++s) {
        const int slot = tid + s * 256;
        const int row  = slot >> 6;           // 64 quads per row
        const int kq   = (slot & 63) << 2;
        async_b128((unsigned)(uintptr_t)&As[sw256(row, kq)],
                   X + (size_t)(rowBase + row) * DEMB + kq);
    }

    // --- B chunk issue: chunk c -> cols (c>>1)*128, k (c&1)*128 -------------
    auto issueB = [&](int c) {
        const int    cb  = (c >> 1) * TILE;
        const int    kb  = (c & 1) * KB;
        float*       buf = Bs + (c & 1) * (TILE * KB);
#pragma unroll
        for (int s = 0; s < 16; ++s) {
            const int slot = tid + s * 256;
            const int row  = slot >> 5;       // 32 quads per row
            const int kq   = (slot & 31) << 2;
            async_b128((unsigned)(uintptr_t)&buf[sw128(row, kq)],
                       X + (size_t)(cb + row) * DEMB + kb + kq);
        }
    };

    issueB(0);                 // outstanding: A(32) + B0(16)
    wait_async_le16();         // in-order completion => A landed
    __syncthreads();

    int rowLab[8];
#pragma unroll
    for (int r = 0; r < 8; ++r) rowLab[r] = labels[rw + r + hf * 8];

    float posBest[8], negBest[8];
    int   posI[8],   negI[8];
#pragma unroll
    for (int r = 0; r < 8; ++r) {
        posBest[r] = -3.402823466e38f; posI[r] = -1;
        negBest[r] =  3.402823466e38f; negI[r] = -1;
    }

    const int rowA = wave * 16 + l15;         // A fragment row (LDS-local)

    for (int jt = 0; jt < B / TILE; ++jt) {
        v8f acc[8] = {};                       // 16x128 output strip

        for (int kc = 0; kc < 2; ++kc) {
            const int c = jt * 2 + kc;
            if (c + 1 < NCHUNK) { issueB(c + 1); wait_async_le16(); }
            else                {                wait_async_le0();  }
            __syncthreads();

            const float* buf   = Bs + (c & 1) * (TILE * KB);
            const int    kbase = kc * KB;
#pragma unroll
            for (int k0 = 0; k0 < KB; k0 += 4) {
                const int ka = k0 + hf * 2;    // lanes 16..31 carry K+2,K+3
                const v2f a = *(const v2f*)&As[sw256(rowA, kbase + ka)];
#pragma unroll
                for (int n = 0; n < 8; ++n) {
                    const v2f b = *(const v2f*)&buf[sw128(n * 16 + l15, ka)];
                    acc[n] = __builtin_amdgcn_wmma_f32_16x16x4_f32(
                        false, a, false, b, (short)0, acc[n], false, false);
                }
            }
            __syncthreads();                   // buffer reusable for next issue
        }

        // Mining update on d2 = sq[col] - 2G (sq[row] const per row; sqrt monotone)
        const int colBase = jt * TILE;
#pragma unroll
        for (int n = 0; n < 8; ++n) {
            const int   col = colBase + n * 16 + l15;
            const int   cl  = labels[col];
            const float sc  = sq[col];
#pragma unroll
            for (int r = 0; r < 8; ++r) {
                const int   row = rw + r + hf * 8;
                const float d2  = sc - 2.0f * acc[n][r];
                if (cl == rowLab[r]) {
                    if (col != row && d2 > posBest[r]) { posBest[r] = d2; posI[r] = col; }
                } else {
                    if (d2 < negBest[r]) { negBest[r] = d2; negI[r] = col; }
                }
            }
        }
    }

    // Cross-lane reduce inside each 16-lane half (offsets < 16 never cross halves).
#pragma unroll
    for (int r = 0; r < 8; ++r) {
        float pv = posBest[r]; int pi = posI[r];
        float nv = negBest[r]; int ni = negI[r];
#pragma unroll
        for (int off = 8; off >= 1; off >>= 1) {
            float opv = __shfl_xor(pv, off, 32); int opi = __shfl_xor(pi, off, 32);
            if (opi >= 0 && (pi < 0 || opv > pv || (opv == pv && opi < pi))) { pv = opv; pi = opi; }
            float onv = __shfl_xor(nv, off, 32); int oni = __shfl_xor(ni, off, 32);
            if (oni >= 0 && (ni < 0 || onv < nv || (onv == nv && oni < ni))) { nv = onv; ni = oni; }
        }
        if (l15 == 0) {
            const int row = rw + r + hf * 8;
            posIdx[row] = pi;
            negIdx[row] = ni;
        }
    }
}

// ---------------------------------------------------------------------------
// Kernel 2: recompute triplet distances (F.pairwise_distance eps-on-diff),
// relu margin, atomic accumulate.
// ---------------------------------------------------------------------------
__global__ __launch_bounds__(256) void loss_kernel(const float* __restrict__ X,
                                                   const int* __restrict__ posIdx,
                                                   const int* __restrict__ negIdx,
                                                   float* __restrict__ sumAcc,
                                                   unsigned* __restrict__ cntAcc) {
    const int a    = blockIdx.x * 8 + (threadIdx.x >> 5);
    const int lane = threadIdx.x & 31;
    const int pi = posIdx[a], ni = negIdx[a];
    const bool valid = (pi >= 0) && (ni >= 0);
    const int pj = valid ? pi : a;
    const int nj = valid ? ni : a;

    const float* xa = X + (size_t)a  * DEMB;
    const float* xp = X + (size_t)pj * DEMB;
    const float* xn = X + (size_t)nj * DEMB;

    float ps = 0.f, ns = 0.f;
#pragma unroll
    for (int d0 = 0; d0 < DEMB / 32; ++d0) {
        const int d = lane + d0 * 32;
        const float dp = xa[d] - xp[d] + PD_EPS; ps += dp * dp;
        const float dn = xa[d] - xn[d] + PD_EPS; ns += dn * dn;
    }
#pragma unroll
    for (int off = 16; off >= 1; off >>= 1) {
        ps += __shfl_xor(ps, off, 32);
        ns += __shfl_xor(ns, off, 32);
    }
    if (lane == 0 && valid) {
        float per = sqrtf(ps) - sqrtf(ns) + MARGIN;
        per = per > 0.f ? per : 0.f;
        atomicAdd(sumAcc, per);
        atomicAdd(cntAcc, 1u);
    }
}

// ---------------------------------------------------------------------------
// Kernel 3: final division.
// ---------------------------------------------------------------------------
__global__ void finalize_kernel(const float* __restrict__ sumAcc,
                                const unsigned* __restrict__ cntAcc,
                                float* __restrict__ out) {
    unsigned c = *cntAcc;
    if (c == 0) c = 1;
    out[0] = *sumAcc / (float)c;
}

// ---------------------------------------------------------------------------
extern "C" void kernel_launch(void* const* d_in, const int* in_sizes, int n_in,
                              void* d_out, int out_size, void* d_ws, size_t ws_size,
                              hipStream_t stream) {
    const float* X      = (const float*)d_in[0];
    const int*   labels = (const int*)d_in[1];
    const int B = in_sizes[1];                 // 8192

    char* ws = (char*)d_ws;
    float*    sq     = (float*)(ws);                       // B floats
    int*      posIdx = (int*)  (ws + (size_t)B * 4);       // B ints
    int*      negIdx = (int*)  (ws + (size_t)B * 8);       // B ints
    float*    sumAcc = (float*)(ws + (size_t)B * 12);
    unsigned* cntAcc = (unsigned*)(ws + (size_t)B * 12 + 4);

    // Allow 256KB dynamic LDS for the mining kernel.
    (void)hipFuncSetAttribute((const void*)mine_kernel,
                              hipFuncAttributeMaxDynamicSharedMemorySize, LDS_BYTES);

    sqnorm_kernel<<<B / 8,    256, 0,         stream>>>(X, sq, sumAcc, cntAcc);
    mine_kernel  <<<B / TILE, 256, LDS_BYTES, stream>>>(X, labels, sq, posIdx, negIdx, B);
    loss_kernel  <<<B / 8,    256, 0,         stream>>>(X, posIdx, negIdx, sumAcc, cntAcc);
    finalize_kernel<<<1, 1, 0, stream>>>(sumAcc, cntAcc, (float*)d_out);
}